// BP_Iteration_11493332484102
// MI455X (gfx1250) — compile-verified
//
#include <hip/hip_runtime.h>
#include <math.h>

// ---- problem constants (match reference) ----
#define N_VARc   16384
#define N_CHKc   8192
#define N_EDGEc  65536
#define BATCHc   1024
#define TPB      256
#define B4       (BATCHc/4)          // 256 float4 per row -> one block per row
#define LLR_CLIPc 15.0f
#define EPSc      1e-6f
#define ABS_MINc  6.1180537e-07f     // -log(tanh(15/2)) in f32

typedef __attribute__((ext_vector_type(4))) float f4;

__device__ __forceinline__ void atomAddF(float* p, float v) {
  unsafeAtomicAdd(p, v);             // hardware global_atomic_add_f32
}

// CDNA5 async global->LDS copy (ASYNCcnt-tracked), one 16B chunk per lane.
// VDST = 32-bit LDS byte address, VADDR = 64-bit global address (GV mode).
__device__ __forceinline__ void async_copy_b128(void* lds_dst, const void* gsrc) {
  unsigned lds_addr = (unsigned)(unsigned long long)lds_dst;  // flat addr[31:0] == LDS offset
  asm volatile("global_load_async_to_lds_b128 %0, %1, off"
               :: "v"(lds_addr), "v"(gsrc)
               : "memory");
}
__device__ __forceinline__ void wait_async0() {
  asm volatile("s_wait_asynccnt 0x0" ::: "memory");
}

// log(tanh(a/2)) = log((1-e^-a)/(1+e^-a)), stable for small a via expm1
__device__ __forceinline__ float logtanh_half(float al) {
  float em = expm1f(-al);            // e^-al - 1  (<= 0)
  return __logf((-em) / (2.0f + em));
}

// ---------------- kernels ----------------

// O[n,b] = Wi * chn_llr[n,b]    (grid = N_VAR*BATCH/4/TPB)
__global__ void k_init_out(const f4* __restrict__ chn, const float* __restrict__ Wip,
                           f4* __restrict__ O) {
  size_t i = (size_t)blockIdx.x * TPB + threadIdx.x;
  float wi = *Wip;
  f4 c = chn[i];
  O[i] = wi * c;
}

__global__ void k_zero(f4* __restrict__ p) {
  size_t i = (size_t)blockIdx.x * TPB + threadIdx.x;
  f4 z = {0.f, 0.f, 0.f, 0.f};
  p[i] = z;
}

// O[vidx[e], b] += We * msg[e, b]     (grid = N_EDGE, block = TPB)
__global__ void k_scatter_var(const f4* __restrict__ msg, const int* __restrict__ vidx,
                              const float* __restrict__ Wep, float* __restrict__ O) {
  int e = blockIdx.x, t = threadIdx.x;
  float we = *Wep;
  int v = vidx[e];                                   // uniform -> scalar load
  f4 m = __builtin_nontemporal_load(&msg[(size_t)e * B4 + t]);
  float* ob = O + (size_t)v * BATCHc + 4 * t;
#pragma unroll
  for (int k = 0; k < 4; ++k) atomAddF(ob + k, we * m[k]);
}

// V-step gather (+ optional fused H-step scatter into amp/sgn accumulators)
// msg_V2C row is staged global->LDS asynchronously; its HBM latency overlaps
// the index scalar load, the O-row gather and the NT stream of msg_C2V.
__global__ void k_vstep(const f4* __restrict__ mC2V, const f4* __restrict__ mV2C,
                        const int* __restrict__ vidx, const int* __restrict__ cidx,
                        const float* __restrict__ Wep, const float* __restrict__ gp,
                        const f4* __restrict__ O, f4* __restrict__ V2Cn,
                        float* __restrict__ amp, unsigned* __restrict__ sgn, int fuse) {
  __shared__ __align__(16) float sbuf[BATCHc];       // staged msg_V2C row (4KB)
  int e = blockIdx.x, t = threadIdx.x;
  size_t row = (size_t)e * B4 + t;
  async_copy_b128(&sbuf[4 * t], &mV2C[row]);         // issue early

  float we = *Wep, g = *gp;
  int v = vidx[e];
  f4 c = __builtin_nontemporal_load(&mC2V[row]);
  f4 s = O[(size_t)v * B4 + t];                      // var accumulator, L2-resident

  wait_async0();                                     // own chunk landed in LDS
  f4 p = *(const f4*)&sbuf[4 * t];

  f4 o;
#pragma unroll
  for (int k = 0; k < 4; ++k) {
    float vn = s[k] - we * c[k];                     // leave-one-out + Wi*chn (in O)
    o[k] = (1.0f - g) * p[k] + g * vn;               // damping
  }
  __builtin_nontemporal_store(o, &V2Cn[row]);
  if (fuse) {
    int cc = cidx[e];
    float*    ab = amp + (size_t)cc * BATCHc + 4 * t;
    unsigned* sb = sgn + (size_t)cc * BATCHc + 4 * t;
#pragma unroll
    for (int k = 0; k < 4; ++k) {
      float lam = fminf(fmaxf(o[k], -LLR_CLIPc), LLR_CLIPc);
      float al  = fminf(fmaxf(fabsf(lam), ABS_MINc), LLR_CLIPc);
      atomAddF(ab + k, logtanh_half(al));
      if (lam < 0.0f) atomicXor(sb + k, 1u);         // parity of negatives
    }
  }
}

// standalone H-step scatter (non-fused fallback path)
__global__ void k_scatter_chk(const f4* __restrict__ V2Cn, const int* __restrict__ cidx,
                              float* __restrict__ amp, unsigned* __restrict__ sgn) {
  int e = blockIdx.x, t = threadIdx.x;
  int cc = cidx[e];
  f4 nv = __builtin_nontemporal_load(&V2Cn[(size_t)e * B4 + t]);
  float*    ab = amp + (size_t)cc * BATCHc + 4 * t;
  unsigned* sb = sgn + (size_t)cc * BATCHc + 4 * t;
#pragma unroll
  for (int k = 0; k < 4; ++k) {
    float lam = fminf(fmaxf(nv[k], -LLR_CLIPc), LLR_CLIPc);
    float al  = fminf(fmaxf(fabsf(lam), ABS_MINc), LLR_CLIPc);
    atomAddF(ab + k, logtanh_half(al));
    if (lam < 0.0f) atomicXor(sb + k, 1u);
  }
}

// H-step gather -> C2V' (+ optional fused M-step scatter into O)
// V2C' row staged async to LDS; overlaps index load + amp/sgn gathers + C2V NT stream.
__global__ void k_hstep(const f4* __restrict__ V2Cn, const f4* __restrict__ mC2V,
                        const int* __restrict__ cidx, const int* __restrict__ vidx,
                        const float* __restrict__ Wep, const float* __restrict__ gp,
                        const float* __restrict__ amp, const unsigned* __restrict__ sgn,
                        f4* __restrict__ C2Vn, float* __restrict__ O, int fuse) {
  __shared__ __align__(16) float sbuf[BATCHc];       // staged V2C' row (4KB)
  int e = blockIdx.x, t = threadIdx.x;
  size_t row = (size_t)e * B4 + t;
  async_copy_b128(&sbuf[4 * t], &V2Cn[row]);         // issue early

  float we = *Wep, g = *gp;
  int cc = cidx[e];
  f4 co = __builtin_nontemporal_load(&mC2V[row]);
  const f4* ab4 = (const f4*)(amp + (size_t)cc * BATCHc) + t;   // L2-resident
  typedef __attribute__((ext_vector_type(4))) unsigned u4;
  const u4* sb4 = (const u4*)(sgn + (size_t)cc * BATCHc) + t;
  f4 av = *ab4;
  u4 sv4 = *sb4;

  wait_async0();
  f4 nv = *(const f4*)&sbuf[4 * t];

  f4 res;
#pragma unroll
  for (int k = 0; k < 4; ++k) {
    float lam  = fminf(fmaxf(nv[k], -LLR_CLIPc), LLR_CLIPc);
    unsigned nb = (lam < 0.0f) ? 1u : 0u;
    float al   = fminf(fmaxf(fabsf(lam), ABS_MINc), LLR_CLIPc);
    float lt   = logtanh_half(al);
    float a    = av[k] - lt;                         // leave-one-out amplitude
    unsigned par = sv4[k] ^ nb;                      // leave-one-out sign parity
    float sv   = (par & 1u) ? -1.0f : 1.0f;
    float x    = __expf(a) * (1.0f - EPSc);
    float h    = sv * __logf((1.0f + x) / (1.0f - x));   // sgn * 2*atanh(exp(a))
    res[k]     = (1.0f - g) * co[k] + g * h;
  }
  __builtin_nontemporal_store(res, &C2Vn[row]);
  if (fuse) {
    int v = vidx[e];
    float* ob = O + (size_t)v * BATCHc + 4 * t;
#pragma unroll
    for (int k = 0; k < 4; ++k) atomAddF(ob + k, we * res[k]);
  }
}

// ---------------- launcher ----------------
extern "C" void kernel_launch(void* const* d_in, const int* in_sizes, int n_in,
                              void* d_out, int out_size, void* d_ws, size_t ws_size,
                              hipStream_t stream) {
  const float* chn  = (const float*)d_in[0];   // (N_VAR, B)
  const float* mC2V = (const float*)d_in[1];   // (E, B)
  const float* mV2C = (const float*)d_in[2];   // (E, B)
  const float* Wi   = (const float*)d_in[3];
  const float* We   = (const float*)d_in[4];
  const float* gm   = (const float*)d_in[5];
  const int*   vidx = (const int*)d_in[6];
  const int*   cidx = (const int*)d_in[7];

  float* C2Vn = (float*)d_out;                                   // (E,B)
  float* V2Cn = C2Vn + (size_t)N_EDGEc * BATCHc;                 // (E,B)
  float* O    = V2Cn + (size_t)N_EDGEc * BATCHc;                 // (N_VAR,B) final output

  const size_t accFloats = (size_t)N_CHKc * BATCHc;              // per accumulator
  const size_t accBytes  = accFloats * sizeof(float) * 2;        // amp + sgn = 64 MiB
  const bool fuse = (ws_size >= accBytes);

  dim3 blk(TPB);
  dim3 gN(N_VARc * BATCHc / (4 * TPB));   // 16384 blocks for (N,B) sweeps
  dim3 gE(N_EDGEc);                       // one block per edge row

  if (fuse) {
    float*    amp = (float*)d_ws;
    unsigned* sgn = (unsigned*)((char*)d_ws + accFloats * sizeof(float));
    k_zero       <<<gN, blk, 0, stream>>>((f4*)d_ws);                         // 64 MiB == N*B floats
    k_init_out   <<<gN, blk, 0, stream>>>((const f4*)chn, Wi, (f4*)O);
    k_scatter_var<<<gE, blk, 0, stream>>>((const f4*)mC2V, vidx, We, O);
    k_vstep      <<<gE, blk, 0, stream>>>((const f4*)mC2V, (const f4*)mV2C, vidx, cidx,
                                          We, gm, (const f4*)O, (f4*)V2Cn, amp, sgn, 1);
    k_init_out   <<<gN, blk, 0, stream>>>((const f4*)chn, Wi, (f4*)O);
    k_hstep      <<<gE, blk, 0, stream>>>((const f4*)V2Cn, (const f4*)mC2V, cidx, vidx,
                                          We, gm, amp, sgn, (f4*)C2Vn, O, 1);
  } else {
    // ws-free fallback: reuse the O region as scratch accumulators between phases
    float*    amp = O;
    unsigned* sgn = (unsigned*)(O + accFloats);
    k_init_out   <<<gN, blk, 0, stream>>>((const f4*)chn, Wi, (f4*)O);
    k_scatter_var<<<gE, blk, 0, stream>>>((const f4*)mC2V, vidx, We, O);
    k_vstep      <<<gE, blk, 0, stream>>>((const f4*)mC2V, (const f4*)mV2C, vidx, cidx,
                                          We, gm, (const f4*)O, (f4*)V2Cn,
                                          (float*)nullptr, (unsigned*)nullptr, 0);
    k_zero       <<<gN, blk, 0, stream>>>((f4*)O);
    k_scatter_chk<<<gE, blk, 0, stream>>>((const f4*)V2Cn, cidx, amp, sgn);
    k_hstep      <<<gE, blk, 0, stream>>>((const f4*)V2Cn, (const f4*)mC2V, cidx, vidx,
                                          We, gm, amp, sgn, (f4*)C2Vn, (float*)nullptr, 0);
    k_init_out   <<<gN, blk, 0, stream>>>((const f4*)chn, Wi, (f4*)O);
    k_scatter_var<<<gE, blk, 0, stream>>>((const f4*)C2Vn, vidx, We, O);
  }
}